// Transformation_81630148428263
// MI455X (gfx1250) — compile-verified
//
#include <hip/hip_runtime.h>
#include <math.h>

// ---------------------------------------------------------------------------
// Bernstein monotonic transform, MI455X / gfx1250 (wave32, WMMA).
//   out[n,v]  = sum_d C(32,d)*p[d,v] * t^d * (1-t)^(32-d)
//   logd[n,v] = log| 32 * sum_d C(31,d)*(p[d+1,v]-p[d,v]) * t^d * (1-t)^(31-d) |
// t = (x[n,v]-lo[v])/(hi[v]-lo[v]);  p = [params[0], cumsum(exp(params[1:]))+params[1]].
//
// Contraction runs on the matrix pipe (V_WMMA_F32_16X16X4_F32) in
// block-diagonal form; VALU computes the t^d / s^e ladders and co-executes.
// ALL intermediates are named scalars (macro-generated straight-line code):
// no source arrays -> SROA trivially promotes, no LDS/scratch traffic.
// ---------------------------------------------------------------------------

#define N_SAMPLES 131072
#define N_VARS    16
#define DEG       32

typedef __attribute__((ext_vector_type(2))) float v2f;
typedef __attribute__((ext_vector_type(8))) float v8f;

// ---------------------------------------------------------------------------
// Prep: fold exp/cumsum restriction + binomials + derivative factor 32 into
// cf[36][16] (forward, zero-padded rows 33..35) and cd[32][16]. Scalar state
// only (no arrays). 16 active lanes, one per variable column.
// ---------------------------------------------------------------------------
__global__ void prep_coeffs_kernel(const float* __restrict__ params,
                                   float* __restrict__ cf,
                                   float* __restrict__ cd) {
  int v = threadIdx.x;
  if (v >= N_VARS) return;

  const float p0    = params[0 * N_VARS + v];
  const float base1 = params[1 * N_VARS + v];

  // forward coefficients: cf[d] = C(32,d) * p[d]
  {
    double b = 1.0;            // C(32,0)
    float  csum = 0.0f;
    cf[0 * N_VARS + v] = (float)b * p0;
    for (int d = 1; d <= DEG; ++d) {
      b = b * (double)(DEG + 1 - d) / (double)d;     // C(32,d)
      csum += expf(params[d * N_VARS + v]);
      const float pd = csum + base1;
      cf[d * N_VARS + v] = (float)b * pd;
    }
    cf[33 * N_VARS + v] = 0.0f;                      // K pad
    cf[34 * N_VARS + v] = 0.0f;
    cf[35 * N_VARS + v] = 0.0f;
  }

  // derivative coefficients: cd[d] = 32 * C(31,d) * (p[d+1]-p[d])
  {
    double b = 1.0;            // C(31,0)
    float  csum = 0.0f;
    float  pd = p0;
    for (int d = 0; d < DEG; ++d) {
      csum += expf(params[(d + 1) * N_VARS + v]);
      const float pn = csum + base1;
      cd[d * N_VARS + v] = 32.0f * (float)b * (pn - pd);
      pd = pn;
      b = b * (double)(DEG - 1 - d) / (double)(d + 1); // C(31,d+1)
    }
  }
}

// ---------------------------------------------------------------------------
// WMMA helpers (macro-generated straight-line code, named scalars only).
// A 16x4 f32 layout: lane = M + 16*(K>>1), vgpr = K&1  -> this lane supplies
// degrees (4c+2*half+q).  B 4x16: lane = N + 16*(K>>1), vgpr = K&1; the block-
// diagonal B is nonzero only in column k (lanes with j==k).
// ---------------------------------------------------------------------------
#define WMMA4(acc, AX, AY, BX, BY)                                            \
  { v2f a_, b_;                                                               \
    a_.x = (AX); a_.y = (AY); b_.x = (BX); b_.y = (BY);                       \
    acc = __builtin_amdgcn_wmma_f32_16x16x4_f32(                              \
        false, a_, false, b_, (short)0, acc, false, false); }

// derivative chunk c over degrees 4c..4c+3 (basis values g0..g3)
#define DCH(c, g0, g1, g2, g3)                                                \
  WMMA4(accD, hsel ? (g2) : (g0), hsel ? (g3) : (g1),                         \
        ksel ? cda##c : 0.0f, ksel ? cdb##c : 0.0f)

// forward chunk c (c=0..7): forward basis = deriv basis * s
#define FCH(c, g0, g1, g2, g3)                                                \
  WMMA4(accF, (hsel ? (g2) : (g0)) * s, (hsel ? (g3) : (g1)) * s,             \
        ksel ? cfa##c : 0.0f, ksel ? cfb##c : 0.0f)

// per-variable body: full scalar ladders + 17 WMMAs
#define BODY(K_, XV_) {                                                       \
  const bool  ksel = (j == (K_));                                             \
  const float lo = pr[(K_)];                                                  \
  const float hi = pr[N_VARS + (K_)];                                         \
  const float t  = ((XV_) - lo) * __builtin_amdgcn_rcpf(hi - lo);             \
  const float s  = 1.0f - t;                                                  \
  /* t powers u_d = t^d */                                                    \
  const float u1 = t,      u2 = u1*t,   u3 = u2*t,   u4 = u3*t;               \
  const float u5 = u4*t,   u6 = u5*t,   u7 = u6*t,   u8 = u7*t;               \
  const float u9 = u8*t,   u10 = u9*t,  u11 = u10*t, u12 = u11*t;             \
  const float u13 = u12*t, u14 = u13*t, u15 = u14*t, u16 = u15*t;             \
  const float u17 = u16*t, u18 = u17*t, u19 = u18*t, u20 = u19*t;             \
  const float u21 = u20*t, u22 = u21*t, u23 = u22*t, u24 = u23*t;             \
  const float u25 = u24*t, u26 = u25*t, u27 = u26*t, u28 = u27*t;             \
  const float u29 = u28*t, u30 = u29*t, u31 = u30*t, t32 = u31*t;             \
  /* s powers q_e = s^e */                                                    \
  const float q1 = s,      q2 = q1*s,   q3 = q2*s,   q4 = q3*s;               \
  const float q5 = q4*s,   q6 = q5*s,   q7 = q6*s,   q8 = q7*s;               \
  const float q9 = q8*s,   q10 = q9*s,  q11 = q10*s, q12 = q11*s;             \
  const float q13 = q12*s, q14 = q13*s, q15 = q14*s, q16 = q15*s;             \
  const float q17 = q16*s, q18 = q17*s, q19 = q18*s, q20 = q19*s;             \
  const float q21 = q20*s, q22 = q21*s, q23 = q22*s, q24 = q23*s;             \
  const float q25 = q24*s, q26 = q25*s, q27 = q26*s, q28 = q27*s;             \
  const float q29 = q28*s, q30 = q29*s, q31 = q30*s;                          \
  /* degree-31 basis w_d = t^d * s^(31-d) (matches reference f32 underflow) */\
  const float w0 = q31,      w1 = u1*q30,   w2 = u2*q29,   w3 = u3*q28;       \
  const float w4 = u4*q27,   w5 = u5*q26,   w6 = u6*q25,   w7 = u7*q24;       \
  const float w8 = u8*q23,   w9 = u9*q22,   w10 = u10*q21, w11 = u11*q20;     \
  const float w12 = u12*q19, w13 = u13*q18, w14 = u14*q17, w15 = u15*q16;     \
  const float w16 = u16*q15, w17 = u17*q14, w18 = u18*q13, w19 = u19*q12;     \
  const float w20 = u20*q11, w21 = u21*q10, w22 = u22*q9,  w23 = u23*q8;      \
  const float w24 = u24*q7,  w25 = u25*q6,  w26 = u26*q5,  w27 = u27*q4;      \
  const float w28 = u28*q3,  w29 = u29*q2,  w30 = u30*q1,  w31 = u31;         \
  /* derivative: 8 WMMAs (K = degrees 0..31 of variable K_) */                \
  DCH(0, w0,  w1,  w2,  w3 )  DCH(1, w4,  w5,  w6,  w7 )                      \
  DCH(2, w8,  w9,  w10, w11)  DCH(3, w12, w13, w14, w15)                      \
  DCH(4, w16, w17, w18, w19)  DCH(5, w20, w21, w22, w23)                      \
  DCH(6, w24, w25, w26, w27)  DCH(7, w28, w29, w30, w31)                      \
  /* forward: 9 WMMAs (degree-32 basis = w_d*s; chunk 8 = degrees 32..35) */  \
  FCH(0, w0,  w1,  w2,  w3 )  FCH(1, w4,  w5,  w6,  w7 )                      \
  FCH(2, w8,  w9,  w10, w11)  FCH(3, w12, w13, w14, w15)                      \
  FCH(4, w16, w17, w18, w19)  FCH(5, w20, w21, w22, w23)                      \
  FCH(6, w24, w25, w26, w27)  FCH(7, w28, w29, w30, w31)                      \
  WMMA4(accF, hsel ? 0.0f : t32, 0.0f,                                        \
        ksel ? cfa8 : 0.0f, ksel ? cfb8 : 0.0f)                               \
}

// straight-line coefficient loads (column j, this lane's degree slots)
#define CLF(c)                                                                \
  const float cfa##c = cfp[(4*(c) + dq0)     * N_VARS + j];                   \
  const float cfb##c = cfp[(4*(c) + dq0 + 1) * N_VARS + j];
#define CLDV(c)                                                               \
  const float cda##c = cdp[(4*(c) + dq0)     * N_VARS + j];                   \
  const float cdb##c = cdp[(4*(c) + dq0 + 1) * N_VARS + j];

// ---------------------------------------------------------------------------
// Main: one wave per 16x16 (samples x variables) tile; 8 waves per block.
// 8192 tiles = 1024 blocks * 8 waves, exact tiling -> EXEC all ones always.
// ---------------------------------------------------------------------------
__launch_bounds__(256)
__global__ void bern_wmma_kernel(const float* __restrict__ x,
                                 const float* __restrict__ pr,    // [2,16]
                                 const float* __restrict__ cfp,   // [36,16]
                                 const float* __restrict__ cdp,   // [32,16]
                                 float* __restrict__ out,
                                 float* __restrict__ logd) {
  const int lane = threadIdx.x & 31;
  const int wave = threadIdx.x >> 5;
  const int tile = blockIdx.x * 8 + wave;
  const int n0   = tile * 16;

  const int  j    = lane & 15;        // D column (variable) == A sample row
  const int  half = lane >> 4;        // K half
  const bool hsel = (half != 0);
  const int  dq0  = half * 2;         // degree offset within a K=4 chunk

  // this lane's sample row: 16 contiguous floats, 64B aligned
  const float4* rp = (const float4*)(x + (size_t)(n0 + j) * N_VARS);
  const float4 r0 = rp[0], r1 = rp[1], r2 = rp[2], r3 = rp[3];
  const float xr0 = r0.x, xr1 = r0.y, xr2 = r0.z, xr3 = r0.w;
  const float xr4 = r1.x, xr5 = r1.y, xr6 = r1.z, xr7 = r1.w;
  const float xr8 = r2.x, xr9 = r2.y, xr10 = r2.z, xr11 = r2.w;
  const float xr12 = r3.x, xr13 = r3.y, xr14 = r3.z, xr15 = r3.w;

  CLF(0) CLF(1) CLF(2) CLF(3) CLF(4) CLF(5) CLF(6) CLF(7) CLF(8)
  CLDV(0) CLDV(1) CLDV(2) CLDV(3) CLDV(4) CLDV(5) CLDV(6) CLDV(7)

  v8f accF = {0.f,0.f,0.f,0.f,0.f,0.f,0.f,0.f};
  v8f accD = {0.f,0.f,0.f,0.f,0.f,0.f,0.f,0.f};

  BODY(0,  xr0 ) BODY(1,  xr1 ) BODY(2,  xr2 ) BODY(3,  xr3 )
  BODY(4,  xr4 ) BODY(5,  xr5 ) BODY(6,  xr6 ) BODY(7,  xr7 )
  BODY(8,  xr8 ) BODY(9,  xr9 ) BODY(10, xr10) BODY(11, xr11)
  BODY(12, xr12) BODY(13, xr13) BODY(14, xr14) BODY(15, xr15)

  // store D fragments: vgpr r holds sample row n0 + r + 8*half, column j
  #pragma unroll
  for (int r = 0; r < 8; ++r) {
    const size_t n = (size_t)(n0 + r + 8 * half);
    out [n * N_VARS + j] = accF[r];
    logd[n * N_VARS + j] = __logf(fabsf(accD[r]));
  }
}

// ---------------------------------------------------------------------------
extern "C" void kernel_launch(void* const* d_in, const int* in_sizes, int n_in,
                              void* d_out, int out_size, void* d_ws, size_t ws_size,
                              hipStream_t stream) {
  const float* x  = (const float*)d_in[0];   // [131072,16] f32
  const float* pm = (const float*)d_in[1];   // [33,16]     f32
  const float* pr = (const float*)d_in[2];   // [2,16]      f32

  float* cf = (float*)d_ws;                  // [36,16]
  float* cd = cf + 36 * N_VARS;              // [32,16]

  float* out  = (float*)d_out;                          // [131072,16]
  float* logd = out + (size_t)N_SAMPLES * N_VARS;       // [131072,16]

  hipLaunchKernelGGL(prep_coeffs_kernel, dim3(1), dim3(16), 0, stream,
                     pm, cf, cd);

  // 8192 tiles of 16 samples; 8 waves (tiles) per 256-thread block.
  hipLaunchKernelGGL(bern_wmma_kernel, dim3(N_SAMPLES / 16 / 8), dim3(256), 0,
                     stream, x, pr, cf, cd, out, logd);
}